// EdgeGATModel_72619307041228
// MI455X (gfx1250) — compile-verified
//
#include <hip/hip_runtime.h>

#define IN_FEATS   128
#define OUT_FEATS  128
#define EDGE_FEATS 64
#define NEG_SLOPE  0.2f

typedef float v2f __attribute__((ext_vector_type(2)));
typedef float v8f __attribute__((ext_vector_type(8)));

// ---- order-preserving float <-> uint map for atomicMax-based segment max ----
__device__ __forceinline__ unsigned ordEnc(float f) {
    unsigned b = __float_as_uint(f);
    return (b & 0x80000000u) ? ~b : (b | 0x80000000u);
}
__device__ __forceinline__ float ordDec(unsigned u) {
    unsigned b = (u & 0x80000000u) ? (u & 0x7FFFFFFFu) : ~u;
    return __uint_as_float(b);
}

// ---------------------------------------------------------------------------
// out[n][c] = bias[c]; maxBits = ord(-inf-ish floor); denom = 0
__global__ __launch_bounds__(256) void init_kernel(float* __restrict__ out,
                                                   const float* __restrict__ bias,
                                                   unsigned* __restrict__ maxBits,
                                                   float* __restrict__ denom, int N) {
    int i = blockIdx.x * blockDim.x + threadIdx.x;
    int total = N * OUT_FEATS;
    if (i < total) out[i] = bias[i & (OUT_FEATS - 1)];
    if (i < N) { maxBits[i] = 0u; denom[i] = 0.f; }
}

// ---------------------------------------------------------------------------
// feat = X @ W via V_WMMA_F32_16X16X4_F32.
// Block = 256 threads = 8 waves. Block covers 16 rows; wave w covers cols [16w,16w+16).
// X tile (16x128) staged in LDS (stride 132 -> conflict-free f32 column reads).
__global__ __launch_bounds__(256) void node_proj_wmma(const float* __restrict__ X,
                                                      const float* __restrict__ W,
                                                      float* __restrict__ feat, int N) {
    __shared__ float xs[16][132];
    const int r0  = blockIdx.x * 16;
    const int tid = threadIdx.x;

    // cooperative stage of 16x128 floats (each thread: 2 x float4)
    for (int j = tid; j < 512; j += 256) {
        int row = j >> 5;
        int col = (j & 31) * 4;
        float4 v = make_float4(0.f, 0.f, 0.f, 0.f);
        if (r0 + row < N)
            v = *(const float4*)&X[(long)(r0 + row) * IN_FEATS + col];
        *(float4*)&xs[row][col] = v;
    }
    __syncthreads();

    const int wave    = tid >> 5;
    const int lane    = tid & 31;
    const int half    = lane >> 4;   // 0: K pair {0,1}, 1: K pair {2,3}
    const int l16     = lane & 15;   // M for A, N for B/D
    const int kOff    = half * 2;
    const int colBase = wave * 16;

    v8f c = {};
    for (int k = 0; k < IN_FEATS; k += 4) {
        const int kk = k + kOff;
        v2f a, b;
        a.x = xs[l16][kk];
        a.y = xs[l16][kk + 1];
        b.x = W[(long)kk       * OUT_FEATS + colBase + l16];
        b.y = W[(long)(kk + 1) * OUT_FEATS + colBase + l16];
        c = __builtin_amdgcn_wmma_f32_16x16x4_f32(
                /*neg_a=*/false, a, /*neg_b=*/false, b,
                /*c_mod=*/(short)0, c, /*reuse_a=*/false, /*reuse_b=*/false);
    }
    // D layout: lanes 0-15 -> M = i, lanes 16-31 -> M = i + 8 (i = VGPR index)
    for (int i = 0; i < 8; ++i) {
        int r = r0 + i + half * 8;
        if (r < N) feat[(long)r * OUT_FEATS + colBase + l16] = c[i];
    }
}

// ---------------------------------------------------------------------------
// el[n] = feat[n,:] . attn_l ; er[n] = feat[n,:] . attn_r   (one wave per node)
__global__ __launch_bounds__(256) void el_er_kernel(const float* __restrict__ feat,
                                                    const float* __restrict__ al,
                                                    const float* __restrict__ ar,
                                                    float* __restrict__ el,
                                                    float* __restrict__ er, int N) {
    int w    = (blockIdx.x * blockDim.x + threadIdx.x) >> 5;
    int lane = threadIdx.x & 31;
    if (w >= N) return;
    float4 f = *(const float4*)&feat[(long)w * OUT_FEATS + lane * 4];
    float4 a = *(const float4*)&al[lane * 4];
    float4 b = *(const float4*)&ar[lane * 4];
    float sl = f.x * a.x + f.y * a.y + f.z * a.z + f.w * a.w;
    float sr = f.x * b.x + f.y * b.y + f.z * b.z + f.w * b.w;
    for (int m = 16; m >= 1; m >>= 1) {
        sl += __shfl_xor(sl, m, 32);
        sr += __shfl_xor(sr, m, 32);
    }
    if (lane == 0) { el[w] = sl; er[w] = sr; }
}

// ---------------------------------------------------------------------------
// wAttnE[j] = sum_k We[j][k] * attn_e[k]   (64x128 mat-vec, one tiny block)
__global__ void we_attn_kernel(const float* __restrict__ We,
                               const float* __restrict__ ae,
                               float* __restrict__ wAttnE) {
    int j = threadIdx.x;
    if (j < EDGE_FEATS) {
        float s = 0.f;
        for (int k = 0; k < OUT_FEATS; ++k) s += We[j * OUT_FEATS + k] * ae[k];
        wAttnE[j] = s;
    }
}

// ---------------------------------------------------------------------------
// per edge: ee = Xe[e,:] . wAttnE ; logit = leaky_relu(el[src]+er[dst]+ee)
// then segment-max via ordered-uint atomicMax. One wave per edge (coalesced 256B row).
__global__ __launch_bounds__(256) void logits_kernel(const float* __restrict__ ef,
                                                     const float* __restrict__ wAttnE,
                                                     const float* __restrict__ el,
                                                     const float* __restrict__ er,
                                                     const int* __restrict__ src,
                                                     const int* __restrict__ dst,
                                                     float* __restrict__ logits,
                                                     unsigned* __restrict__ maxBits, int E) {
    int e    = (blockIdx.x * blockDim.x + threadIdx.x) >> 5;
    int lane = threadIdx.x & 31;
    if (e >= E) return;
    float2 f  = *(const float2*)&ef[(long)e * EDGE_FEATS + lane * 2];
    float2 wv = *(const float2*)&wAttnE[lane * 2];
    float s = f.x * wv.x + f.y * wv.y;
    for (int m = 16; m >= 1; m >>= 1) s += __shfl_xor(s, m, 32);
    if (lane == 0) {
        float l = el[src[e]] + er[dst[e]] + s;
        l = (l > 0.f) ? l : NEG_SLOPE * l;
        logits[e] = l;
        atomicMax(&maxBits[dst[e]], ordEnc(l));
    }
}

// ---------------------------------------------------------------------------
// alpha = exp(logit - segmax[dst]) (stored over logits); denom[dst] += alpha
__global__ __launch_bounds__(256) void alpha_kernel(float* __restrict__ logits,
                                                    const unsigned* __restrict__ maxBits,
                                                    const int* __restrict__ dst,
                                                    float* __restrict__ denom, int E) {
    int e = blockIdx.x * blockDim.x + threadIdx.x;
    if (e >= E) return;
    int d = dst[e];
    float a = expf(logits[e] - ordDec(maxBits[d]));
    logits[e] = a;
    atomicAdd(&denom[d], a);
}

// ---------------------------------------------------------------------------
// out[dst,:] += feat[src,:] * (alpha / denom[dst])  (wave per edge, 4 f32 atomics/lane)
__global__ __launch_bounds__(256) void scatter_kernel(const float* __restrict__ feat,
                                                      const float* __restrict__ alpha,
                                                      const float* __restrict__ denom,
                                                      const int* __restrict__ src,
                                                      const int* __restrict__ dst,
                                                      float* __restrict__ out, int E) {
    int e    = (blockIdx.x * blockDim.x + threadIdx.x) >> 5;
    int lane = threadIdx.x & 31;
    if (e >= E) return;
    int s = src[e], d = dst[e];
    float a = 0.f;
    if (lane == 0) {
        float dn = denom[d];
        a = alpha[e] / ((dn == 0.f) ? 1.f : dn);
    }
    a = __shfl(a, 0, 32);
    float4 f = *(const float4*)&feat[(long)s * OUT_FEATS + lane * 4];
    float* o = &out[(long)d * OUT_FEATS + lane * 4];
    atomicAdd(o + 0, f.x * a);
    atomicAdd(o + 1, f.y * a);
    atomicAdd(o + 2, f.z * a);
    atomicAdd(o + 3, f.w * a);
}

// ---------------------------------------------------------------------------
extern "C" void kernel_launch(void* const* d_in, const int* in_sizes, int n_in,
                              void* d_out, int out_size, void* d_ws, size_t ws_size,
                              hipStream_t stream) {
    const float* Xn   = (const float*)d_in[0];
    const float* Xe   = (const float*)d_in[1];
    const int*   src  = (const int*)d_in[2];
    const int*   dst  = (const int*)d_in[3];
    const float* W    = (const float*)d_in[4];
    const float* We   = (const float*)d_in[5];
    const float* al   = (const float*)d_in[6];
    const float* ar   = (const float*)d_in[7];
    const float* ae   = (const float*)d_in[8];
    const float* bias = (const float*)d_in[9];

    const int N = in_sizes[0] / IN_FEATS;
    const int E = in_sizes[2];

    // workspace carve-out (256B aligned)
    char*  ws  = (char*)d_ws;
    size_t off = 0;
    auto carve = [&](size_t bytes) -> void* {
        void* p = ws + off;
        off += (bytes + 255) & ~size_t(255);
        return p;
    };
    float*    feat    = (float*)carve((size_t)N * OUT_FEATS * sizeof(float));
    float*    el      = (float*)carve((size_t)N * sizeof(float));
    float*    er      = (float*)carve((size_t)N * sizeof(float));
    float*    wAttnE  = (float*)carve(EDGE_FEATS * sizeof(float));
    float*    logits  = (float*)carve((size_t)E * sizeof(float));  // reused as alpha
    unsigned* maxBits = (unsigned*)carve((size_t)N * sizeof(unsigned));
    float*    denom   = (float*)carve((size_t)N * sizeof(float));

    float* out = (float*)d_out;

    int initBlocks = (N * OUT_FEATS + 255) / 256;
    init_kernel<<<initBlocks, 256, 0, stream>>>(out, bias, maxBits, denom, N);

    we_attn_kernel<<<1, 64, 0, stream>>>(We, ae, wAttnE);

    node_proj_wmma<<<(N + 15) / 16, 256, 0, stream>>>(Xn, W, feat, N);

    el_er_kernel<<<(N + 7) / 8, 256, 0, stream>>>(feat, al, ar, el, er, N);

    logits_kernel<<<(E + 7) / 8, 256, 0, stream>>>(Xe, wAttnE, el, er, src, dst,
                                                   logits, maxBits, E);

    alpha_kernel<<<(E + 255) / 256, 256, 0, stream>>>(logits, maxBits, dst, denom, E);

    scatter_kernel<<<(E + 7) / 8, 256, 0, stream>>>(feat, logits, denom, src, dst, out, E);
}